// TwoLayerSimpleHeteroGAT_5265629905487
// MI455X (gfx1250) — compile-verified
//
#include <hip/hip_runtime.h>
#include <math.h>

// ---------------------------------------------------------------------------
// Two-layer hetero GAT (user<->item), MI455X / gfx1250, wave32.
// Dense GEMMs: V_WMMA_F32_16X16X4_F32, one wave per 16-row stripe covering all
// column tiles (1 global A load -> CT WMMAs), B staged transposed in LDS so
// each B fragment is a single conflict-free ds_load_b64.
// Edge softmax/aggregation: f32 global atomics (the memory-bound part).
// ---------------------------------------------------------------------------

#define NUSR 100000
#define NITM 100000
#define EMBK 64
#define DD   32      // D1 == D2
#define HH   3
#define FF   96      // H*D
#define EE   500000
#define SLOPE 0.01f
#define EPSBN 1e-5f

typedef __attribute__((ext_vector_type(2))) float v2f;
typedef __attribute__((ext_vector_type(8))) float v8f;

// ---------------- WMMA fp32 GEMM: C[M,N] = A[M,K] @ B[K,N] -----------------
// CT = N/16 column tiles, all owned by one wave (16-row stripe per wave).
// mode 0: store raw
// mode 1: + bias[col]                       (residual-projection init of rst)
// mode 2: + bias[col], then BN(eval)+ReLU   (the Linear->BN->ReLU blocks)
// Requirements: M %16 == 0, K %4 == 0, K <= 96, CT*16 == N <= 96.
template <int CT>
__global__ __launch_bounds__(256) void gemm_wmma_f32(
    const float* __restrict__ A, const float* __restrict__ Bg,
    float* __restrict__ C, int M, int K,
    const float* __restrict__ bias, const float* __restrict__ bn, int mode)
{
  constexpr int N = CT * 16;
  const int Ks = K + 2;                        // padded LDS stride (bank-conflict free)
  __shared__ float Bs[96 * 98];                // transposed: Bs[col*Ks + k]
  for (int i = threadIdx.x; i < K * N; i += blockDim.x) {
    int k = i / N, c = i % N;                  // Bg is row-major [K, N]
    Bs[c * Ks + k] = Bg[i];
  }
  __syncthreads();

  const int rowTiles = M >> 4;
  const int wid = blockIdx.x * (blockDim.x >> 5) + (threadIdx.x >> 5);
  if (wid >= rowTiles) return;                 // wave-uniform exit

  const int lane = threadIdx.x & 31;
  const int half = lane >> 4;                  // 0: lanes 0-15, 1: lanes 16-31
  const int lrow = lane & 15;

  const float* Arow = A + (size_t)((wid << 4) + lrow) * K;

  v8f acc[CT];
#pragma unroll
  for (int t = 0; t < CT; ++t) acc[t] = (v8f){};

  for (int k0 = 0; k0 < K; k0 += 4) {
    const int ka = k0 + (half << 1);           // A 16x4: v0/v1 = K 0,1 (lo) / 2,3 (hi)
    v2f a = *(const v2f*)(Arow + ka);          // one global b64 feeds CT WMMAs
#pragma unroll
    for (int t = 0; t < CT; ++t) {             // B 4x16: rows ka, ka+1 striped on lanes
      v2f b = *(const v2f*)(&Bs[(t * 16 + lrow) * Ks + ka]);  // ds_load_b64
      acc[t] = __builtin_amdgcn_wmma_f32_16x16x4_f32(false, a, false, b,
                                                     (short)0, acc[t], false, false);
    }
  }

#pragma unroll
  for (int t = 0; t < CT; ++t) {
    const int col = t * 16 + lrow;
    float add = (mode >= 1) ? bias[col] : 0.0f;
    float g = 1.f, be = 0.f, mu = 0.f, iv = 1.f;
    if (mode == 2) {                           // bn layout: [gamma,beta,mean,var][N]
      g  = bn[0 * N + col];
      be = bn[1 * N + col];
      mu = bn[2 * N + col];
      iv = rsqrtf(bn[3 * N + col] + EPSBN);
    }
#pragma unroll
    for (int j = 0; j < 8; ++j) {              // C vgpr j: row j (lo half) / j+8 (hi)
      const int row = (wid << 4) + j + (half << 3);
      float y = acc[t][j] + add;
      if (mode == 2) { y = (y - mu) * g * iv + be; y = fmaxf(y, 0.0f); }
      C[(size_t)row * N + col] = y;
    }
  }
}

// ------------- el/er: out[n,h] = sum_d feat[n,h*32+d] * attn[h,d] ----------
__global__ void attn_dot_kernel(const float* __restrict__ feat,
                                const float* __restrict__ attn,
                                float* __restrict__ out, int n)
{
  int idx = blockIdx.x * blockDim.x + threadIdx.x;
  if (idx >= n * HH) return;
  int node = idx / HH, h = idx % HH;
  const float* f = feat + (size_t)node * FF + h * DD;
  const float* a = attn + h * DD;
  float s = 0.f;
#pragma unroll
  for (int d = 0; d < DD; ++d) s += f[d] * a[d];
  out[idx] = s;
}

__global__ void fill_kernel(float* __restrict__ p, float v, int n)
{
  int i = blockIdx.x * blockDim.x + threadIdx.x;
  if (i < n) p[i] = v;
}

// float atomic max via monotone int/uint bit patterns
__device__ __forceinline__ void atomicMaxF(float* addr, float v)
{
  if (v >= 0.0f) atomicMax((int*)addr, __float_as_int(v));
  else           atomicMin((unsigned int*)addr, __float_as_uint(v));
}

// per (edge,head): e = leaky(el[src]+er[dst]); segment max over dst
__global__ void edge_max_kernel(const float* __restrict__ el,
                                const float* __restrict__ er,
                                const int* __restrict__ src,
                                const int* __restrict__ dst,
                                float* __restrict__ ebuf,
                                float* __restrict__ m)
{
  int idx = blockIdx.x * blockDim.x + threadIdx.x;
  if (idx >= EE * HH) return;
  int e = idx / HH, h = idx % HH;
  float s = el[src[e] * HH + h] + er[dst[e] * HH + h];
  s = s > 0.0f ? s : SLOPE * s;
  ebuf[idx] = s;
  atomicMaxF(&m[dst[e] * HH + h], s);
}

// per (edge,head): ex = exp(e - m[dst]); segment sum over dst
__global__ void edge_exp_kernel(const int* __restrict__ dst,
                                float* __restrict__ ebuf,
                                const float* __restrict__ m,
                                float* __restrict__ den)
{
  int idx = blockIdx.x * blockDim.x + threadIdx.x;
  if (idx >= EE * HH) return;
  int e = idx / HH, h = idx % HH;
  float ex = expf(ebuf[idx] - m[dst[e] * HH + h]);
  ebuf[idx] = ex;
  atomicAdd(&den[dst[e] * HH + h], ex);
}

// one wave per edge: rst[dst] += fs[src] * (ex / den[dst]), 96 floats/edge
__global__ __launch_bounds__(256) void edge_scatter_kernel(
    const float* __restrict__ fs, const float* __restrict__ ebuf,
    const float* __restrict__ den, const int* __restrict__ src,
    const int* __restrict__ dst, float* __restrict__ rst)
{
  int wid = blockIdx.x * (blockDim.x >> 5) + (threadIdx.x >> 5);
  if (wid >= EE) return;                       // wave-uniform
  int lane = threadIdx.x & 31;
  int s = src[wid], t = dst[wid];
  const float* fr = fs + (size_t)s * FF;
  float* rr = rst + (size_t)t * FF;
  __builtin_prefetch(rr + lane, 1, 0);         // warm dst row (global_prefetch_b8)
#pragma unroll
  for (int h = 0; h < HH; ++h) {
    float c = ebuf[wid * HH + h] / den[t * HH + h];
    atomicAdd(&rr[h * DD + lane], fr[h * DD + lane] * c);
  }
}

// ---------------------------------------------------------------------------
static inline int cdiv(int a, int b) { return (a + b - 1) / b; }

static void launch_gemm(hipStream_t st, const float* A, const float* B, float* C,
                        int M, int K, int N, const float* bias, const float* bn,
                        int mode)
{
  int blocks = cdiv(M / 16, 8);                // one wave per 16-row stripe
  if (N == 96)
    gemm_wmma_f32<6><<<blocks, 256, 0, st>>>(A, B, C, M, K, bias, bn, mode);
  else                                         // N == 32
    gemm_wmma_f32<2><<<blocks, 256, 0, st>>>(A, B, C, M, K, bias, bn, mode);
}

// One GAT relation pass: hsrc --(src->dst edges)--> writes RST[Ndst, 96]
static void run_gat(hipStream_t st,
                    const float* hsrc, const float* hdst,
                    int Nsrc, int Ndst, int K,
                    const float* W, const float* attnL, const float* attnR,
                    const float* resW, const float* bias,
                    const int* src, const int* dst,
                    float* FS, float* FD, float* EL, float* ER,
                    float* Mb, float* DEN, float* EB, float* RST)
{
  launch_gemm(st, hsrc, W,    FS,  Nsrc, K, FF, nullptr, nullptr, 0);  // fs
  launch_gemm(st, hdst, W,    FD,  Ndst, K, FF, nullptr, nullptr, 0);  // fd
  launch_gemm(st, hdst, resW, RST, Ndst, K, FF, bias,    nullptr, 1);  // res + bias init

  attn_dot_kernel<<<cdiv(Nsrc * HH, 256), 256, 0, st>>>(FS, attnL, EL, Nsrc);
  attn_dot_kernel<<<cdiv(Ndst * HH, 256), 256, 0, st>>>(FD, attnR, ER, Ndst);

  fill_kernel<<<cdiv(Ndst * HH, 256), 256, 0, st>>>(Mb, -INFINITY, Ndst * HH);
  fill_kernel<<<cdiv(Ndst * HH, 256), 256, 0, st>>>(DEN, 0.0f, Ndst * HH);

  edge_max_kernel<<<cdiv(EE * HH, 256), 256, 0, st>>>(EL, ER, src, dst, EB, Mb);
  edge_exp_kernel<<<cdiv(EE * HH, 256), 256, 0, st>>>(dst, EB, Mb, DEN);
  edge_scatter_kernel<<<cdiv(EE, 8), 256, 0, st>>>(FS, EB, DEN, src, dst, RST);
}

extern "C" void kernel_launch(void* const* d_in, const int* in_sizes, int n_in,
                              void* d_out, int out_size, void* d_ws, size_t ws_size,
                              hipStream_t stream)
{
  const float* emb_user = (const float*)d_in[0];
  const float* emb_item = (const float*)d_in[1];
  const float* W1    = (const float*)d_in[2];    // [2, 64, 96]
  const float* attn1 = (const float*)d_in[3];    // [2, 2, 3, 32]
  const float* res1  = (const float*)d_in[4];    // [2, 64, 96]
  const float* bias1 = (const float*)d_in[5];    // [2, 96]
  const float* nnW1  = (const float*)d_in[6];    // [2, 96, 32]
  const float* nnb1  = (const float*)d_in[7];    // [2, 32]
  const float* bn1   = (const float*)d_in[8];    // [2, 4, 32]
  const float* W2    = (const float*)d_in[9];    // [2, 32, 96]
  const float* attn2 = (const float*)d_in[10];
  const float* res2  = (const float*)d_in[11];   // [2, 32, 96]
  const float* bias2 = (const float*)d_in[12];
  const float* nnW2  = (const float*)d_in[13];
  const float* nnb2  = (const float*)d_in[14];
  const float* bn2   = (const float*)d_in[15];
  const int* go_src   = (const int*)d_in[16];
  const int* go_dst   = (const int*)d_in[17];
  const int* back_src = (const int*)d_in[18];
  const int* back_dst = (const int*)d_in[19];

  float* ws = (float*)d_ws;
  size_t o = 0;
  auto carve = [&](size_t n) { float* p = ws + o; o += n; return p; };
  float* FS    = carve((size_t)NUSR * FF);
  float* FD    = carve((size_t)NUSR * FF);
  float* RST_I = carve((size_t)NITM * FF);
  float* RST_U = carve((size_t)NUSR * FF);
  float* EL    = carve((size_t)NUSR * HH);
  float* ER    = carve((size_t)NUSR * HH);
  float* Mb    = carve((size_t)NUSR * HH);
  float* DEN   = carve((size_t)NUSR * HH);
  float* EB    = carve((size_t)EE * HH);
  float* HU1   = carve((size_t)NUSR * DD);
  float* HI1   = carve((size_t)NITM * DD);
  (void)ws_size; (void)n_in; (void)in_sizes; (void)out_size;

  const int HD = HH * DD;  // 96

  // ---- Layer 1 ----
  // rel 0 'go': user -> item
  run_gat(stream, emb_user, emb_item, NUSR, NITM, EMBK,
          W1 + 0, attn1 + (0 * 2 + 0) * HD, attn1 + (0 * 2 + 1) * HD,
          res1 + 0, bias1 + 0, go_src, go_dst,
          FS, FD, EL, ER, Mb, DEN, EB, RST_I);
  // rel 1 'back': item -> user
  run_gat(stream, emb_item, emb_user, NITM, NUSR, EMBK,
          W1 + (size_t)EMBK * FF, attn1 + (1 * 2 + 0) * HD, attn1 + (1 * 2 + 1) * HD,
          res1 + (size_t)EMBK * FF, bias1 + FF, back_src, back_dst,
          FS, FD, EL, ER, Mb, DEN, EB, RST_U);
  // Linear -> BN -> ReLU (type 0 = user, type 1 = item)
  launch_gemm(stream, RST_U, nnW1 + 0,               HU1, NUSR, FF, DD, nnb1 + 0,  bn1 + 0,      2);
  launch_gemm(stream, RST_I, nnW1 + (size_t)FF * DD, HI1, NITM, FF, DD, nnb1 + DD, bn1 + 4 * DD, 2);

  // ---- Layer 2 ----
  run_gat(stream, HU1, HI1, NUSR, NITM, DD,
          W2 + 0, attn2 + (0 * 2 + 0) * HD, attn2 + (0 * 2 + 1) * HD,
          res2 + 0, bias2 + 0, go_src, go_dst,
          FS, FD, EL, ER, Mb, DEN, EB, RST_I);
  run_gat(stream, HI1, HU1, NITM, NUSR, DD,
          W2 + (size_t)DD * FF, attn2 + (1 * 2 + 0) * HD, attn2 + (1 * 2 + 1) * HD,
          res2 + (size_t)DD * FF, bias2 + FF, back_src, back_dst,
          FS, FD, EL, ER, Mb, DEN, EB, RST_U);

  float* out_user = (float*)d_out;                       // [NU, 32]
  float* out_item = (float*)d_out + (size_t)NUSR * DD;   // [NI, 32]
  launch_gemm(stream, RST_U, nnW2 + 0,               out_user, NUSR, FF, DD, nnb2 + 0,  bn2 + 0,      2);
  launch_gemm(stream, RST_I, nnW2 + (size_t)FF * DD, out_item, NITM, FF, DD, nnb2 + DD, bn2 + 4 * DD, 2);
}